// GraphAttentionLayer_53541062312138
// MI455X (gfx1250) — compile-verified
//
#include <hip/hip_runtime.h>
#include <hip/hip_bf16.h>

#define GAT_N      8192
#define GAT_FIN    128
#define GAT_FOUT   64
#define GAT_ALPHA  0.2f
#define GAT_NEGINF (-9.0e15f)
#define LOG2E      1.4426950408889634f

typedef __attribute__((ext_vector_type(2))) float v2f;
typedef __attribute__((ext_vector_type(8))) float v8f;

// D = A(16x4 f32) * B(4x16 f32) + C(16x16 f32)
#define WMMA_F32_16x16x4(A_, B_, C_) \
  __builtin_amdgcn_wmma_f32_16x16x4_f32(false, (A_), false, (B_), (short)0, (C_), false, false)

// ---------------------------------------------------------------------------
// Kernel 1: Wh = h @ W  (8192x64), s_src = Wh@a1, s_dst = Wh@a2
// One block per row i, 64 threads (one per output feature).
// ---------------------------------------------------------------------------
__global__ __launch_bounds__(64) void gat_wh_kernel(
    const float* __restrict__ h, const float* __restrict__ W,
    const float* __restrict__ a,
    float* __restrict__ Wh, float* __restrict__ ssrc, float* __restrict__ sdst) {
  const int i = blockIdx.x;
  const int t = threadIdx.x;  // 0..63
  const float* __restrict__ hrow = h + (size_t)i * GAT_FIN;

  float acc = 0.0f;
#pragma unroll 8
  for (int k = 0; k < GAT_FIN; ++k)
    acc = fmaf(hrow[k], W[k * GAT_FOUT + t], acc);
  Wh[(size_t)i * GAT_FOUT + t] = acc;

  __shared__ float red1[64];
  __shared__ float red2[64];
  red1[t] = acc * a[t];
  red2[t] = acc * a[GAT_FOUT + t];
  __syncthreads();
#pragma unroll
  for (int s = 32; s > 0; s >>= 1) {
    if (t < s) { red1[t] += red1[t + s]; red2[t] += red2[t + s]; }
    __syncthreads();
  }
  if (t == 0) { ssrc[i] = red1[0]; sdst[i] = red2[0]; }
}

// ---------------------------------------------------------------------------
// Kernel 2: fused masked-softmax attention + (attention @ Wh) + ELU.
// 512 blocks (16 rows each) x 128 threads (4 waves). Wave w owns j-range
// [w*2048, (w+1)*2048). Pass 1 computes per-row (max, sumexp); pass 2 does
// the P @ Wh product with V_WMMA_F32_16X16X4_F32 (slab re-read hits L2).
// ---------------------------------------------------------------------------
__global__ __launch_bounds__(128) void gat_attn_kernel(
    const int* __restrict__ adj, const float* __restrict__ Wh,
    const float* __restrict__ ssrc, const float* __restrict__ sdst,
    float* __restrict__ out) {
  const int i0   = blockIdx.x * 16;
  const int tid  = threadIdx.x;
  const int wave = tid >> 5;   // 0..3
  const int lane = tid & 31;
  const int r    = lane & 15;  // row within 16-row block (also N-col within tile)
  const int g    = lane >> 4;  // lane half (K split for f32 wmma fragments)
  const int row  = i0 + r;

  __shared__ float m_part[4][16];
  __shared__ float l_part[4][16];
  __shared__ float acc_lds[4][16][GAT_FOUT];

  const float sr = ssrc[row];
  const int jq0 = wave * 2048;

  // ---------------- Pass 1: softmax stats over this wave's j-quarter -------
  {
    float m = -3.0e38f;
    float l = 0.0f;
    const int j0 = jq0 + g * 1024;  // lane pair (l, l+16) splits the quarter
    const int*   __restrict__ arow = adj + (size_t)row * GAT_N + j0;
    const float* __restrict__ sdp  = sdst + j0;
#pragma unroll 2
    for (int jj = 0; jj < 1024; jj += 4) {
      const int4   av = *(const int4*)(arow + jj);
      const float4 dv = *(const float4*)(sdp + jj);
      float ev[4], s;
      s = sr + dv.x; s = s > 0.f ? s : GAT_ALPHA * s; ev[0] = av.x > 0 ? s : GAT_NEGINF;
      s = sr + dv.y; s = s > 0.f ? s : GAT_ALPHA * s; ev[1] = av.y > 0 ? s : GAT_NEGINF;
      s = sr + dv.z; s = s > 0.f ? s : GAT_ALPHA * s; ev[2] = av.z > 0 ? s : GAT_NEGINF;
      s = sr + dv.w; s = s > 0.f ? s : GAT_ALPHA * s; ev[3] = av.w > 0 ? s : GAT_NEGINF;
#pragma unroll
      for (int q = 0; q < 4; ++q) {
        const float e = ev[q];
        if (e > m) { l = l * __builtin_amdgcn_exp2f((m - e) * LOG2E) + 1.0f; m = e; }
        else       { l += __builtin_amdgcn_exp2f((e - m) * LOG2E); }
      }
    }
    // merge the two lanes (g=0 / g=1) that share a row
    const float mo = __shfl_xor(m, 16, 32);
    const float lo = __shfl_xor(l, 16, 32);
    const float M2 = fmaxf(m, mo);
    const float L2 = l  * __builtin_amdgcn_exp2f((m  - M2) * LOG2E) +
                     lo * __builtin_amdgcn_exp2f((mo - M2) * LOG2E);
    if (g == 0) { m_part[wave][r] = M2; l_part[wave][r] = L2; }
  }
  __syncthreads();

  // merged row max across the 4 wave-partials (uniform across lane pair)
  float Mr = -3.0e38f;
#pragma unroll
  for (int w = 0; w < 4; ++w) Mr = fmaxf(Mr, m_part[w][r]);

  // ---------------- Pass 2: P @ Wh via f32 WMMA over j-quarter -------------
  v8f cacc[4] = {v8f{}, v8f{}, v8f{}, v8f{}};  // 16 rows x 64 cols accumulator
  const int* __restrict__ arow2 = adj + (size_t)row * GAT_N;

  for (int jt = jq0; jt < jq0 + 2048; jt += 16) {
    // Build A fragments (P tile, 16x16) in wmma 16x4 layout:
    // wmma step t covers K = jt+4t .. jt+4t+3; lane half g holds K = +2g, +2g+1
    v2f A[4];
#pragma unroll
    for (int t = 0; t < 4; ++t) {
      const int k0 = jt + 4 * t + 2 * g;
      const int2   av = *(const int2*)(arow2 + k0);
      const float2 dv = *(const float2*)(sdst + k0);
      float s0 = sr + dv.x; s0 = s0 > 0.f ? s0 : GAT_ALPHA * s0;
      float s1 = sr + dv.y; s1 = s1 > 0.f ? s1 : GAT_ALPHA * s1;
      const float e0 = av.x > 0 ? s0 : GAT_NEGINF;
      const float e1 = av.y > 0 ? s1 : GAT_NEGINF;
      A[t].x = __builtin_amdgcn_exp2f((e0 - Mr) * LOG2E);
      A[t].y = __builtin_amdgcn_exp2f((e1 - Mr) * LOG2E);
    }
    // B fragments from Wh rows (K) x 16-col N-tiles, then 16 wmma
#pragma unroll
    for (int t = 0; t < 4; ++t) {
      const int k0 = jt + 4 * t + 2 * g;
      const float* __restrict__ b0 = Wh + (size_t)k0 * GAT_FOUT + r;
      const float* __restrict__ b1 = b0 + GAT_FOUT;
#pragma unroll
      for (int nt = 0; nt < 4; ++nt) {
        v2f B;
        B.x = b0[nt * 16];
        B.y = b1[nt * 16];
        cacc[nt] = WMMA_F32_16x16x4(A[t], B, cacc[nt]);
      }
    }
  }

  // dump C fragments: vgpr rr -> row rr (lanes 0-15) / 8+rr (lanes 16-31)
#pragma unroll
  for (int nt = 0; nt < 4; ++nt) {
#pragma unroll
    for (int rr = 0; rr < 8; ++rr) {
      const int orow = g == 0 ? rr : 8 + rr;
      acc_lds[wave][orow][nt * 16 + r] = cacc[nt][rr];
    }
  }
  __syncthreads();

  // ---------------- Finalize: sum split-j partials, 1/L, ELU ---------------
  for (int idx = tid; idx < 16 * GAT_FOUT; idx += 128) {
    const int orow = idx >> 6;
    const int ocol = idx & 63;
    const float ssum = acc_lds[0][orow][ocol] + acc_lds[1][orow][ocol] +
                       acc_lds[2][orow][ocol] + acc_lds[3][orow][ocol];
    float M = -3.0e38f;
#pragma unroll
    for (int w = 0; w < 4; ++w) M = fmaxf(M, m_part[w][orow]);
    float L = 0.0f;
#pragma unroll
    for (int w = 0; w < 4; ++w)
      L += l_part[w][orow] * __builtin_amdgcn_exp2f((m_part[w][orow] - M) * LOG2E);
    const float hp = ssum / L;
    out[(size_t)(i0 + orow) * GAT_FOUT + ocol] = hp > 0.0f ? hp : expm1f(hp);
  }
}

// ---------------------------------------------------------------------------
extern "C" void kernel_launch(void* const* d_in, const int* in_sizes, int n_in,
                              void* d_out, int out_size, void* d_ws, size_t ws_size,
                              hipStream_t stream) {
  const float* h   = (const float*)d_in[0];
  const int*   adj = (const int*)d_in[1];
  const float* W   = (const float*)d_in[2];
  const float* a   = (const float*)d_in[3];
  float* out = (float*)d_out;

  float* Wh   = (float*)d_ws;                 // 8192*64 floats
  float* ssrc = Wh + (size_t)GAT_N * GAT_FOUT;  // 8192 floats
  float* sdst = ssrc + GAT_N;                   // 8192 floats

  gat_wh_kernel<<<GAT_N, 64, 0, stream>>>(h, W, a, Wh, ssrc, sdst);
  gat_attn_kernel<<<GAT_N / 16, 128, 0, stream>>>(adj, Wh, ssrc, sdst, out);
}